// SumLayer_31851477467726
// MI455X (gfx1250) — compile-verified
//
#include <hip/hip_runtime.h>
#include <math.h>

// Segmented log-sum-exp over sorted segment ids (CSR-style).
//   out[s] = log( sum_{e: csr[e]==s} exp(x[ptrs[e]] - m_s) + 1e-15 ) + m_s
// One thread per segment; block stages its whole edge window into LDS via
// gfx1250 async global->LDS copies (ASYNCcnt), then two in-LDS passes.

#define TPB 256      // 8 wave32s per workgroup
#define CAP 8192     // max staged edges per block (32KB LDS); mean is 4096

#if __has_builtin(__builtin_amdgcn_global_load_async_to_lds_b32)
#define HAVE_ASYNC_LDS 1
typedef int __attribute__((address_space(1))) gint_t;   // global
typedef int __attribute__((address_space(3))) lint_t;   // LDS
#endif

__device__ __forceinline__ int lower_bound(const int* __restrict__ a, int n, int key) {
    int lo = 0, hi = n;
    while (lo < hi) {
        int mid = (lo + hi) >> 1;
        if (a[mid] < key) lo = mid + 1; else hi = mid;
    }
    return lo;
}

__global__ __launch_bounds__(TPB)
void seg_lse_kernel(const float* __restrict__ x,
                    const int*   __restrict__ ptrs,
                    const int*   __restrict__ csr,
                    float*       __restrict__ out,
                    int E, int nseg) {
    __shared__ int sbeg[TPB + 1];
    __shared__ int buf[CAP];        // holds edge ptrs, then gathered float bits

    const int s0 = blockIdx.x * TPB;
    const int s  = s0 + threadIdx.x;

    // One binary search per thread; neighbor supplies the upper bound.
    // (keys beyond nseg simply return E, which is harmless)
    int beg = lower_bound(csr, E, s);
    sbeg[threadIdx.x] = beg;
    if (threadIdx.x == 0) sbeg[TPB] = lower_bound(csr, E, s0 + TPB);
    __syncthreads();

    const int blockBeg = sbeg[0];
    const int nEdges   = sbeg[TPB] - blockBeg;
    const int end      = sbeg[threadIdx.x + 1];

    if (nEdges <= CAP) {
        // ---- Stage the block's edge window into LDS, coalesced ----
#ifdef HAVE_ASYNC_LDS
        for (int i = threadIdx.x; i < nEdges; i += TPB) {
            __builtin_amdgcn_global_load_async_to_lds_b32(
                (gint_t*)(const_cast<int*>(ptrs) + blockBeg + i),
                (lint_t*)(buf + i),
                /*offset=*/0, /*cpol=*/0);
        }
        asm volatile("s_wait_asynccnt 0" ::: "memory");
#else
        for (int i = threadIdx.x; i < nEdges; i += TPB) {
            buf[i] = ptrs[blockBeg + i];
        }
#endif
        __syncthreads();

        if (s < nseg) {
            const int lbeg = beg - blockBeg;
            const int lend = end - blockBeg;
            // Pass 1: gather x once (L2-resident), track max, cache value in LDS.
            // Each edge slot is owned by exactly one thread -> in-place overwrite is safe.
            float m = -__builtin_inff();
            for (int i = lbeg; i < lend; ++i) {
                int   p = buf[i];
                float v = x[p];
                buf[i]  = __float_as_int(v);
                m = fmaxf(m, v);
            }
            // Pass 2: sum exp(v - m) from LDS-cached values.
            float sum = 0.0f;
            for (int i = lbeg; i < lend; ++i) {
                sum += expf(__int_as_float(buf[i]) - m);
            }
            out[s] = logf(sum + 1e-15f) + m;
        }
    } else {
        // Fallback (block-uniform, astronomically rare): straight from global.
        if (s < nseg) {
            float m = -__builtin_inff();
            for (int e = beg; e < end; ++e) {
                m = fmaxf(m, x[ptrs[e]]);
            }
            float sum = 0.0f;
            for (int e = beg; e < end; ++e) {
                sum += expf(x[ptrs[e]] - m);
            }
            out[s] = logf(sum + 1e-15f) + m;
        }
    }
}

extern "C" void kernel_launch(void* const* d_in, const int* in_sizes, int n_in,
                              void* d_out, int out_size, void* d_ws, size_t ws_size,
                              hipStream_t stream) {
    const float* x    = (const float*)d_in[0];
    const int*   ptrs = (const int*)  d_in[1];
    const int*   csr  = (const int*)  d_in[2];
    float*       out  = (float*)      d_out;

    const int E    = in_sizes[1];   // number of edges
    const int nseg = out_size;      // number of output segments

    const int nblocks = (nseg + TPB - 1) / TPB;
    seg_lse_kernel<<<nblocks, TPB, 0, stream>>>(x, ptrs, csr, out, E, nseg);
}